// NB_3221225472037
// MI455X (gfx1250) — compile-verified
//
#include <hip/hip_runtime.h>
#include <math.h>

#define VOCAB 50257
#define NW    1571            // ceil(VOCAB/32) bitmap words per column
#define NWP   1572            // padded LDS stride (conflict-free)
#define KPAD  (NW * 32)       // 50272 padded K extent
#define SEQ   400
#define BATCH 2048
#define CPB   16              // batch columns per block
#define THREADS 256
#define NWAVES  8             // THREADS / 32 (wave32)

#define WS_SCALARS 4                          // logZ0, logZ1, logc0, logc1
#define WS_NEEDED  (sizeof(float) * (WS_SCALARS + (size_t)KPAD * 4))

typedef float v2f __attribute__((ext_vector_type(2)));
typedef float v8f __attribute__((ext_vector_type(8)));

// ---------------------------------------------------------------------------
// Prep A: logZ[y] = log(sum_v xycounts[v,y]), logc[y] = log(ycounts[y]).
// One block; fixed partition + fixed tree => deterministic.
// ---------------------------------------------------------------------------
__global__ __launch_bounds__(THREADS)
void nb_prep(const float* __restrict__ xyc, const float* __restrict__ yc,
             float* __restrict__ ws) {
    __shared__ float r0[THREADS], r1[THREADS];
    float s0 = 0.f, s1 = 0.f;
    for (int v = threadIdx.x; v < VOCAB; v += THREADS) {
        s0 += xyc[2 * v + 0];
        s1 += xyc[2 * v + 1];
    }
    r0[threadIdx.x] = s0;
    r1[threadIdx.x] = s1;
    __syncthreads();
    for (int off = THREADS / 2; off > 0; off >>= 1) {
        if ((int)threadIdx.x < off) {
            r0[threadIdx.x] += r0[threadIdx.x + off];
            r1[threadIdx.x] += r1[threadIdx.x + off];
        }
        __syncthreads();
    }
    if (threadIdx.x == 0) {
        ws[0] = logf(r0[0]);
        ws[1] = logf(r1[0]);
        ws[2] = logf(yc[0]);
        ws[3] = logf(yc[1]);
    }
}

// ---------------------------------------------------------------------------
// Prep B: padded B-table bt[k][4] = { log xyc[k,0], log xyc[k,1], 1.0, 0.0 }
// (zeros past VOCAB). Slot 2 = ones column (n_uniq); slot 3 = 0 for pad lanes.
// ---------------------------------------------------------------------------
__global__ __launch_bounds__(THREADS)
void nb_fill(const float* __restrict__ xyc, float4* __restrict__ bt) {
    const int stride = gridDim.x * THREADS;
    for (int v = blockIdx.x * THREADS + threadIdx.x; v < KPAD; v += stride) {
        float4 e;
        if (v < VOCAB) {
            e.x = logf(xyc[2 * v + 0]);
            e.y = logf(xyc[2 * v + 1]);
            e.z = 1.0f;
        } else {
            e.x = 0.0f; e.y = 0.0f; e.z = 0.0f;
        }
        e.w = 0.0f;
        bt[v] = e;
    }
}

// ---------------------------------------------------------------------------
// Main: 16 columns per block.
//   Phase 1: zero 16 vocab bitmaps (~100 KB of the 320 KB WGP LDS).
//   Phase 2: ds_or_b32 presence marking (idempotent -> deterministic).
//   Phase 3: D += presence(16xK) x B(Kx16) via v_wmma_f32_16x16x4_f32.
//            USE_BT: branchless clamped loads from bt (no log in loop).
//            else : on-the-fly __logf fallback.
//   Phase 4: fixed-order cross-wave reduce, emit scores.
// ---------------------------------------------------------------------------
template <bool USE_BT>
__global__ __launch_bounds__(THREADS)
void nb_score(const int* __restrict__ x, const float* __restrict__ xyc,
              const float* __restrict__ bt, const float* __restrict__ ws,
              float* __restrict__ out) {
    __shared__ unsigned bm[CPB][NWP];
    __shared__ float red[NWAVES][CPB][4];

    const int tid  = threadIdx.x;
    const int colb = blockIdx.x * CPB;

    // ---- Phase 1: zero bitmaps ----
    unsigned* bmf = &bm[0][0];
    for (int i = tid; i < CPB * NWP; i += THREADS) bmf[i] = 0u;
    __syncthreads();

    // ---- Phase 2: mark presence ----
    for (int i = tid; i < CPB * SEQ; i += THREADS) {
        int c = i & (CPB - 1);
        int s = i >> 4;
        int tok = x[s * BATCH + colb + c];
        atomicOr(&bm[c][tok >> 5], 1u << (tok & 31));
    }
    __syncthreads();

    // ---- Phase 3: WMMA K-sweep ----
    const int lane = tid & 31;
    const int wave = tid >> 5;
    const int col  = lane & 15;          // A-row M == B-col N for this lane
    const int half = lane >> 4;          // selects K rows {k0+2h, k0+2h+1}
    const int coff = (col < 4) ? col : 3;        // clamp: slot 3 holds 0.0f
    const int poff = 8 * half + coff;            // lane-invariant float offset

    v8f acc = {};

    const int wpw  = (NW + NWAVES - 1) / NWAVES;     // 197 words per wave
    const int wbeg = wave * wpw;
    const int wend = (wbeg + wpw < NW) ? (wbeg + wpw) : NW;

    for (int w = wbeg; w < wend; ++w) {
        const unsigned bits = bm[col][w];
        const int kbase = w << 5;
        #pragma unroll
        for (int j = 0; j < 8; ++j) {
            v2f a, b;
            const int bsh = 4 * j + 2 * half;
            a.x = (float)((bits >> bsh) & 1u);
            a.y = (float)((bits >> (bsh + 1)) & 1u);
            if (USE_BT) {
                // bt is [KPAD][4]; row krow = kbase + 4j + 2*half
                const int fidx = 4 * kbase + 16 * j + poff;
                b.x = bt[fidx];          // row krow,   col coff
                b.y = bt[fidx + 4];      // row krow+1, col coff
            } else {
                const int krow = kbase + 4 * j + 2 * half;
                if (col < 2) {
                    float p0 = (krow     < VOCAB) ? xyc[2 * krow       + col] : 1.0f;
                    float p1 = (krow + 1 < VOCAB) ? xyc[2 * (krow + 1) + col] : 1.0f;
                    b.x = __logf(p0);
                    b.y = __logf(p1);
                } else if (col == 2) {
                    b.x = 1.0f; b.y = 1.0f;
                } else {
                    b.x = 0.0f; b.y = 0.0f;
                }
            }
            acc = __builtin_amdgcn_wmma_f32_16x16x4_f32(
                false, a, false, b, (short)0, acc, false, false);
        }
    }

    // ---- Phase 4: reduce across waves (fixed order) ----
    if (col < 3) {
        #pragma unroll
        for (int r = 0; r < 8; ++r) {
            // D layout: VGPR r, lanes 0-15 -> M=r ; lanes 16-31 -> M=r+8
            red[wave][r + 8 * half][col] = acc[r];
        }
    }
    __syncthreads();

    if (tid < 32) {
        int m = tid >> 1;                // column within block
        int y = tid & 1;                 // class
        float s = 0.f, n = 0.f;
        for (int wv = 0; wv < NWAVES; ++wv) {
            s += red[wv][m][y];
            n += red[wv][m][2];
        }
        out[(colb + m) * 2 + y] = s - n * ws[y] + ws[2 + y];
    }
}

// ---------------------------------------------------------------------------
extern "C" void kernel_launch(void* const* d_in, const int* in_sizes, int n_in,
                              void* d_out, int out_size, void* d_ws, size_t ws_size,
                              hipStream_t stream) {
    const int*   x   = (const int*)d_in[0];     // [SEQ, BATCH] int32
    const float* xyc = (const float*)d_in[1];   // [VOCAB, 2] f32
    const float* yc  = (const float*)d_in[2];   // [2] f32
    float* out = (float*)d_out;                 // [BATCH, 2] f32
    float* ws  = (float*)d_ws;

    nb_prep<<<1, THREADS, 0, stream>>>(xyc, yc, ws);

    if (ws_size >= WS_NEEDED) {
        float* bt = ws + WS_SCALARS;
        nb_fill<<<128, THREADS, 0, stream>>>(xyc, (float4*)bt);
        nb_score<true><<<BATCH / CPB, THREADS, 0, stream>>>(x, xyc, bt, ws, out);
    } else {
        nb_score<false><<<BATCH / CPB, THREADS, 0, stream>>>(x, xyc, nullptr, ws, out);
    }
}